// MLPDecoder_69896297775695
// MI455X (gfx1250) — compile-verified
//
#include <hip/hip_runtime.h>

typedef __attribute__((ext_vector_type(2))) float v2f;
typedef __attribute__((ext_vector_type(8))) float v8f;

#define NNODES  100000
#define DIM     128
#define HIDDEN  128
#define NEDGE   640000
#define KTOT    256              // 2*DIM (concat of src,dst rows)
#define KSTEPS  64               // KTOT / 4 (K per WMMA = 4)
#define NTILES  8                // HIDDEN / 16
#define WAVES   8
#define REPS    4                // iterations per wave; 2 M-tiles (32 edges) each
#define EDGES_PER_BLOCK (WAVES * REPS * 32)          // 1024
#define BLOCKS_PER_SIGN (NEDGE / EDGES_PER_BLOCK)    // 625

static __device__ __forceinline__ v8f wmma4(v2f a, v2f b, v8f c) {
    return __builtin_amdgcn_wmma_f32_16x16x4_f32(false, a, false, b,
                                                 (short)0, c, false, false);
}

__global__ __launch_bounds__(256) void mlp_edge_decoder_kernel(
    const float* __restrict__ z,
    const int*   __restrict__ ei_pos,
    const int*   __restrict__ ei_neg,
    const float* __restrict__ W1,
    const float* __restrict__ b1,
    const float* __restrict__ W2,
    const float* __restrict__ b2,
    float*       __restrict__ out)
{
    // W1 pre-swizzled into WMMA B-fragment order: 256*128 floats = 128 KB LDS
    __shared__ float sW1[KTOT * HIDDEN];

    const int tid  = threadIdx.x;
    const int lane = tid & 31;
    const int wave = tid >> 5;

    const int sign = blockIdx.x / BLOCKS_PER_SIGN;   // 0 = pos, 1 = neg
    const int blk  = blockIdx.x % BLOCKS_PER_SIGN;
    const int* __restrict__ ei = sign ? ei_neg : ei_pos;
    float* __restrict__ outp = out + (size_t)sign * NEDGE;

    // ---- Swizzle W1 [256,128] row-major into B-fragment order ----
    // element e = ((kstep*8 + ntile)*32 + lane)*2 + j
    //   -> W1[(kstep*4 + 2*(lane>>4) + j) * 128 + ntile*16 + (lane&15)]
    for (int e = tid; e < KTOT * HIDDEN; e += 256) {
        int j    = e & 1;
        int l    = (e >> 1) & 31;
        int frag = e >> 6;
        int nt   = frag & 7;
        int ks   = frag >> 3;
        int k    = ks * 4 + ((l >> 4) << 1) + j;
        int n    = nt * 16 + (l & 15);
        sW1[e] = W1[k * HIDDEN + n];
    }
    __syncthreads();

    const int nl    = lane & 15;            // N (column) slot within a tile
    const int khalf = (lane >> 4) << 1;     // K sub-offset (+0 or +2) for A/B frags
    const int pfoff = (lane >> 4) << 6;     // prefetch offset: lane<16 -> +0B, >=16 -> +256B

    // per-lane second-layer weights and biases (L2-resident broadcasts)
    float w2v[NTILES];
    float b1v[NTILES];
#pragma unroll
    for (int t = 0; t < NTILES; ++t) {
        w2v[t] = W2[t * 16 + nl];
        b1v[t] = b1[t * 16 + nl];
    }
    const float b2v = b2[0];

    const float* fragBase = &sW1[lane << 1];  // + frag*64 floats

    for (int rep = 0; rep < REPS; ++rep) {
        const int ebase = blk * EDGES_PER_BLOCK + wave * (REPS * 32) + rep * 32;
        // two M-tiles per wave: rows ebase..+15 and ebase+16..+31
        const int e0 = ebase + nl;
        const int e1 = ebase + 16 + nl;
        const long s0 = (long)ei[e0];
        const long d0 = (long)ei[NEDGE + e0];
        const long s1 = (long)ei[e1];
        const long d1 = (long)ei[NEDGE + e1];
        const float* __restrict__ sr0 = z + s0 * (long)DIM + khalf;
        const float* __restrict__ dr0 = z + d0 * (long)DIM + khalf;
        const float* __restrict__ sr1 = z + s1 * (long)DIM + khalf;
        const float* __restrict__ dr1 = z + d1 * (long)DIM + khalf;

        // prefetch next rep's gathered rows into cache while we compute
        if (rep + 1 < REPS) {
            const int ne0 = ebase + 32 + nl;
            const int ne1 = ebase + 48 + nl;
            __builtin_prefetch(z + (long)ei[ne0] * DIM + pfoff, 0, 3);
            __builtin_prefetch(z + (long)ei[NEDGE + ne0] * DIM + pfoff, 0, 3);
            __builtin_prefetch(z + (long)ei[ne1] * DIM + pfoff, 0, 3);
            __builtin_prefetch(z + (long)ei[NEDGE + ne1] * DIM + pfoff, 0, 3);
        }

        // accumulators init with layer-1 bias (same for every row M)
        v8f acc0[NTILES], acc1[NTILES];
#pragma unroll
        for (int t = 0; t < NTILES; ++t) {
            v8f c;
#pragma unroll
            for (int j = 0; j < 8; ++j) c[j] = b1v[t];
            acc0[t] = c;
            acc1[t] = c;
        }

        // ---- K = 0..127 : src-node half of the concat ----
#pragma unroll 2
        for (int ks = 0; ks < KSTEPS / 2; ++ks) {
            v2f a0 = *(const v2f*)(sr0 + ks * 4);
            v2f a1 = *(const v2f*)(sr1 + ks * 4);
            const float* fb = fragBase + (ks << 9);    // ks * 8 frags * 64 floats
#pragma unroll
            for (int t = 0; t < NTILES; ++t) {
                v2f bfr = *(const v2f*)(fb + (t << 6));   // one LDS read, two WMMAs
                acc0[t] = wmma4(a0, bfr, acc0[t]);
                acc1[t] = wmma4(a1, bfr, acc1[t]);
            }
        }
        // ---- K = 128..255 : dst-node half of the concat ----
#pragma unroll 2
        for (int ks = 0; ks < KSTEPS / 2; ++ks) {
            v2f a0 = *(const v2f*)(dr0 + ks * 4);
            v2f a1 = *(const v2f*)(dr1 + ks * 4);
            const float* fb = fragBase + ((ks + KSTEPS / 2) << 9);
#pragma unroll
            for (int t = 0; t < NTILES; ++t) {
                v2f bfr = *(const v2f*)(fb + (t << 6));
                acc0[t] = wmma4(a0, bfr, acc0[t]);
                acc1[t] = wmma4(a1, bfr, acc1[t]);
            }
        }

        // ---- Layer 2: score = relu(h) @ W2 + b2, for both M-tiles ----
        // acc[t][j] = h[row (j + 8*(lane>=16)), col t*16+nl]
        float p0[8], p1[8];
#pragma unroll
        for (int j = 0; j < 8; ++j) { p0[j] = 0.f; p1[j] = 0.f; }
#pragma unroll
        for (int t = 0; t < NTILES; ++t) {
#pragma unroll
            for (int j = 0; j < 8; ++j) {
                float h0 = acc0[t][j];
                float h1 = acc1[t][j];
                p0[j] += (h0 > 0.f ? h0 : 0.f) * w2v[t];
                p1[j] += (h1 > 0.f ? h1 : 0.f) * w2v[t];
            }
        }
        // reduce over the 16 lanes of each half-wave (xor 1,2,4,8 stays in-half)
#pragma unroll
        for (int off = 1; off < 16; off <<= 1) {
#pragma unroll
            for (int j = 0; j < 8; ++j) {
                p0[j] += __shfl_xor(p0[j], off, 32);
                p1[j] += __shfl_xor(p1[j], off, 32);
            }
        }
        if (nl == 0) {
            const int mbase = (lane >> 4) * 8;   // lane 0 -> rows 0..7, lane 16 -> rows 8..15
#pragma unroll
            for (int j = 0; j < 8; ++j) {
                outp[ebase + mbase + j]      = p0[j] + b2v;
                outp[ebase + 16 + mbase + j] = p1[j] + b2v;
            }
        }
    }
}

extern "C" void kernel_launch(void* const* d_in, const int* in_sizes, int n_in,
                              void* d_out, int out_size, void* d_ws, size_t ws_size,
                              hipStream_t stream) {
    const float* z      = (const float*)d_in[0];
    const int*   ei_pos = (const int*)  d_in[1];
    const int*   ei_neg = (const int*)  d_in[2];
    const float* W1     = (const float*)d_in[3];
    const float* b1     = (const float*)d_in[4];
    const float* W2     = (const float*)d_in[5];
    const float* b2     = (const float*)d_in[6];
    float* out = (float*)d_out;

    dim3 grid(2 * BLOCKS_PER_SIGN);   // 1250 blocks: first half pos, second half neg
    dim3 block(256);                  // 8 waves (wave32)
    mlp_edge_decoder_kernel<<<grid, block, 0, stream>>>(
        z, ei_pos, ei_neg, W1, b1, W2, b2, out);
}